// SIR_GNN_32066225832565
// MI455X (gfx1250) — compile-verified
//
#include <hip/hip_runtime.h>

#define B_ 64
#define T_ 168
#define N_ 400

typedef __attribute__((ext_vector_type(2))) float v2f;
typedef __attribute__((ext_vector_type(8))) float v8f;

// ---------------------------------------------------------------------------
// Kernel A: phi = softmax_j(relu(phi_p*adj)) * adj ; rowsum[i]; Np[i]=phi@pop
// One block per row i (400 blocks, 128 threads).
// ---------------------------------------------------------------------------
__global__ void __launch_bounds__(128)
phi_rowstats_kernel(const float* __restrict__ phi_p,
                    const float* __restrict__ adj,
                    const float* __restrict__ pop,
                    float* __restrict__ phi,
                    float* __restrict__ rowsum,
                    float* __restrict__ nparr) {
  __shared__ float vals[N_];
  __shared__ float red0[128];
  __shared__ float red1[128];
  const int i   = blockIdx.x;
  const int tid = threadIdx.x;
  const float* prow = phi_p + i * N_;
  const float* arow = adj   + i * N_;

  // relu(phi_p*adj), row max (softmax over ALL 400 entries, incl. zeros)
  float lmax = 0.0f;
  for (int j = tid; j < N_; j += 128) {
    float x = prow[j] * arow[j];
    x = x > 0.0f ? x : 0.0f;
    vals[j] = x;
    lmax = fmaxf(lmax, x);
  }
  red0[tid] = lmax;
  __syncthreads();
  for (int s = 64; s > 0; s >>= 1) {
    if (tid < s) red0[tid] = fmaxf(red0[tid], red0[tid + s]);
    __syncthreads();
  }
  const float m = red0[0];
  __syncthreads();

  // denominator
  float lsum = 0.0f;
  for (int j = tid; j < N_; j += 128) lsum += __expf(vals[j] - m);
  red0[tid] = lsum;
  __syncthreads();
  for (int s = 64; s > 0; s >>= 1) {
    if (tid < s) red0[tid] += red0[tid + s];
    __syncthreads();
  }
  const float rdenom = 1.0f / red0[0];
  __syncthreads();

  // phi, rowsum, Np
  float lrow = 0.0f, lnp = 0.0f;
  for (int j = tid; j < N_; j += 128) {
    float p = __expf(vals[j] - m) * rdenom * arow[j];
    phi[i * N_ + j] = p;
    lrow += p;
    lnp  += p * pop[j];
  }
  red0[tid] = lrow;
  red1[tid] = lnp;
  __syncthreads();
  for (int s = 64; s > 0; s >>= 1) {
    if (tid < s) { red0[tid] += red0[tid + s]; red1[tid] += red1[tid + s]; }
    __syncthreads();
  }
  if (tid == 0) {
    rowsum[i] = red0[0];
    nparr[i]  = red1[0];
  }
}

// ---------------------------------------------------------------------------
// Kernel B: fused argmin + suffix-sum over T in ONE pass of input (17.2 MB),
// vectorized: each lane owns 4 consecutive columns -> global_load_b128
// (16 B/lane, fully coalesced; 4x fewer VMEM ops on the dominant stream).
// Produces S, x_last, and w[b,k] = beta[k]/Np[k] * S[b,k] * x_last[b,k].
// Also copies `ind` bit-exact into the output tail.
// ---------------------------------------------------------------------------
__global__ void __launch_bounds__(128)
suffix_stats_kernel(const float* __restrict__ input,
                    const int*   __restrict__ ind,
                    const float* __restrict__ beta,
                    const float* __restrict__ gama,
                    const float* __restrict__ pop,
                    const float* __restrict__ nparr,
                    float* __restrict__ w,
                    float* __restrict__ xlast,
                    int*   __restrict__ ind_out) {
  const int b   = blockIdx.x;
  const int tid = threadIdx.x;
  if (tid == 0) ind_out[b] = ind[b];          // passthrough second output
  if (tid >= N_ / 4) return;                  // 100 active lanes
  const int n0 = tid * 4;

  const float4* base = (const float4*)(input + (size_t)b * T_ * N_ + n0);
  float4 v      = base[0];
  float4 curmin = v;
  float4 ssum   = v;                           // suffix sum from (first) argmin
  float4 xl     = v;
  for (int t = 1; t < T_; ++t) {
    __builtin_prefetch((const void*)(base + (size_t)(t + 6) * (N_ / 4)), 0, 1);
    v = base[(size_t)t * (N_ / 4)];
    // strict '<' == first-occurrence argmin; reset suffix sum on new min
    if (v.x < curmin.x) { curmin.x = v.x; ssum.x = v.x; } else { ssum.x += v.x; }
    if (v.y < curmin.y) { curmin.y = v.y; ssum.y = v.y; } else { ssum.y += v.y; }
    if (v.z < curmin.z) { curmin.z = v.z; ssum.z = v.z; } else { ssum.z += v.z; }
    if (v.w < curmin.w) { curmin.w = v.w; ssum.w = v.w; } else { ssum.w += v.w; }
    xl = v;                                    // ends at t = T-1
  }
  const float g1 = 1.0f + gama[0];
  float4 wv, Sv;
  Sv.x = pop[n0 + 0] - ssum.x * g1;
  Sv.y = pop[n0 + 1] - ssum.y * g1;
  Sv.z = pop[n0 + 2] - ssum.z * g1;
  Sv.w = pop[n0 + 3] - ssum.w * g1;
  wv.x = (beta[n0 + 0] / nparr[n0 + 0]) * Sv.x * xl.x;
  wv.y = (beta[n0 + 1] / nparr[n0 + 1]) * Sv.y * xl.y;
  wv.z = (beta[n0 + 2] / nparr[n0 + 2]) * Sv.z * xl.z;
  wv.w = (beta[n0 + 3] / nparr[n0 + 3]) * Sv.w * xl.w;
  ((float4*)(w     + b * N_ + n0))[0] = wv;
  ((float4*)(xlast + b * N_ + n0))[0] = xl;
}

// ---------------------------------------------------------------------------
// Kernel C: out[b,i] = rowsum[i] * (sum_k phi[i,k]*w[b,k]) + xlast[b,i]
// via V_WMMA_F32_16X16X4_F32. One wave per 16x16 output tile:
//   M = batch tile (4 tiles of 16 over B=64), N = output-col tile (25 tiles),
//   K loops over 400 in steps of 4 (100 WMMAs per tile).
// A frag (16x4 f32, 2 VGPR): lanes 0-15 -> K=0,1; lanes 16-31 -> K=2,3; M=lane&15.
// B frag (4x16 f32, 2 VGPR): mirrored; B[k,n] = phi[n,k] (phi^T).
// C/D: standard 8-VGPR 16x16 layout (VGPR r: M=r / r+8; N=lane&15).
// ---------------------------------------------------------------------------
__global__ void __launch_bounds__(32)
wmma_out_kernel(const float* __restrict__ w,
                const float* __restrict__ phi,
                const float* __restrict__ rowsum,
                const float* __restrict__ xlast,
                float* __restrict__ out) {
  const int tile  = blockIdx.x;     // 0..99
  const int mtile = tile & 3;       // 4 tiles over B
  const int ntile = tile >> 2;      // 25 tiles over N
  const int lane  = threadIdx.x;
  const int half  = lane >> 4;      // 0 or 1
  const int l15   = lane & 15;
  const int b0    = mtile * 16;
  const int n0    = ntile * 16;
  const int kb    = half * 2;       // K sub-offset held by this half-wave

  const float* arow = w   + (b0 + l15) * N_;  // A row m = l15
  const float* brow = phi + (n0 + l15) * N_;  // B col n = l15 (phi^T access)

  v8f acc = {};
  for (int k0 = 0; k0 < N_; k0 += 4) {
    v2f a, b;
    a.x = arow[k0 + kb];
    a.y = arow[k0 + kb + 1];
    b.x = brow[k0 + kb];
    b.y = brow[k0 + kb + 1];
    // (neg_a, A, neg_b, B, c_mod, C, reuse_a, reuse_b)
    acc = __builtin_amdgcn_wmma_f32_16x16x4_f32(false, a, false, b,
                                                (short)0, acc, false, false);
  }

  const int   nidx = n0 + l15;
  const float rs   = rowsum[nidx];
#pragma unroll
  for (int r = 0; r < 8; ++r) {
    const int m    = r + half * 8;
    const int bidx = b0 + m;
    out[bidx * N_ + nidx] = rs * acc[r] + xlast[bidx * N_ + nidx];
  }
}

// ---------------------------------------------------------------------------
// Workspace layout (floats):
//   phi    : [0      , 160000)   400x400
//   rowsum : [160000 , 160400)
//   Np     : [160400 , 160800)
//   w      : [160800 , 186400)   64x400
//   xlast  : [186400 , 212000)   64x400
// ---------------------------------------------------------------------------
extern "C" void kernel_launch(void* const* d_in, const int* in_sizes, int n_in,
                              void* d_out, int out_size, void* d_ws, size_t ws_size,
                              hipStream_t stream) {
  const float* input = (const float*)d_in[0];
  const int*   ind   = (const int*)  d_in[1];
  const float* phi_p = (const float*)d_in[2];
  const float* beta  = (const float*)d_in[3];
  const float* gama  = (const float*)d_in[4];
  const float* adj   = (const float*)d_in[5];
  const float* pop   = (const float*)d_in[6];

  float* out = (float*)d_out;
  float* ws  = (float*)d_ws;
  float* phi    = ws;
  float* rowsum = ws + 160000;
  float* nparr  = ws + 160400;
  float* w      = ws + 160800;
  float* xlast  = ws + 186400;
  int*   indout = (int*)(out + B_ * N_);   // tuple output #2, bit-exact i32

  phi_rowstats_kernel<<<N_, 128, 0, stream>>>(phi_p, adj, pop, phi, rowsum, nparr);
  suffix_stats_kernel<<<B_, 128, 0, stream>>>(input, ind, beta, gama, pop, nparr,
                                              w, xlast, indout);
  wmma_out_kernel<<<100, 32, 0, stream>>>(w, phi, rowsum, xlast, out);
}